// raindrop_fuse_68805376082361
// MI455X (gfx1250) — compile-verified
//
#include <hip/hip_runtime.h>

typedef __attribute__((ext_vector_type(2))) float v2f;
typedef __attribute__((ext_vector_type(8))) float v8f;

#define NIMG 16
#define HIMG 100
#define WIMG 252
#define CFEAT 64
#define TILE 16

// t1-tile LDS offset for im2col index k = (ky*3+kx)*16 + cin  (t1 tile is 20x20x16)
__host__ __device__ constexpr int boff_of(int k) {
    return (((k >> 4) / 3) * 20 + ((k >> 4) % 3)) * 16 + (k & 15);
}

// ---------------------------------------------------------------------------
__global__ void zero_loss_kernel(float* loss) { *loss = 0.f; }

// ---------------------------------------------------------------------------
// Fused conv1(2->16,relu) -> conv2(16->32,relu via WMMA f32 16x16x4) ->
// conv3(32->2) on flow_map, producing flow (N,H,W,2) + smooth_l1 loss sum.
// One block = one 16x16 output tile; all intermediates LDS-resident.
// ---------------------------------------------------------------------------
__global__ __launch_bounds__(256) void flow_conv_kernel(
    const float* __restrict__ flow_map,   // (N,H,W,2)
    const float* __restrict__ flow_gt,    // (2N,H,W)
    const float* __restrict__ w1, const float* __restrict__ b1,   // (16,2,3,3)
    const float* __restrict__ w2, const float* __restrict__ b2,   // (32,16,3,3)
    const float* __restrict__ w3, const float* __restrict__ b3,   // (2,32,3,3)
    float* __restrict__ flow_out,         // (N,H,W,2)
    float* __restrict__ loss_acc)
{
    extern __shared__ float smem[];
    float* w2ld = smem;                 // 4608  : repacked [k/2=72][m=32][2]
    float* t1ld = w2ld + 4608;          // 6400  : 20x20 positions x 16 ch
    float* t2ld = t1ld + 6400;          // 10752 : 336 positions x 32 ch
    float* inld = t2ld + 10752;         // 968   : 22x22 positions x 2 ch
    float* w1ld = inld + 968;           // 288
    float* b1ld = w1ld + 288;           // 16
    float* b2ld = b1ld + 16;            // 32
    float* w3ld = b2ld + 32;            // 576
    float* b3ld = w3ld + 576;           // 2
    float* red  = b3ld + 2;             // 256

    const int tid = threadIdx.x;
    const int n   = blockIdx.z;
    const int h0  = blockIdx.y * TILE;
    const int w0  = blockIdx.x * TILE;

    // ---- stage weights ----
    for (int i = tid; i < 288; i += 256) w1ld[i] = w1[i];
    for (int i = tid; i < 576; i += 256) w3ld[i] = w3[i];
    if (tid < 16) b1ld[tid] = b1[tid];
    if (tid < 32) b2ld[tid] = b2[tid];
    if (tid < 2)  b3ld[tid] = b3[tid];
    // repack W2: dest[(k>>1)*64 + m*2 + (k&1)] = w2[m][cin][ky][kx], k=(ky*3+kx)*16+cin
    for (int i = tid; i < 4608; i += 256) {
        int k = i >> 5, m = i & 31;
        int cin = k & 15, tap = k >> 4;
        int ky = tap / 3, kx = tap - 3 * ky;
        w2ld[(k >> 1) * 64 + m * 2 + (k & 1)] = w2[((m * 16 + cin) * 3 + ky) * 3 + kx];
    }
    // ---- stage zero-padded input patch rows h0-3..h0+18, cols w0-3..w0+18 ----
    for (int i = tid; i < 22 * 22 * 2; i += 256) {
        int c = i & 1, pos = i >> 1;
        int r = pos / 22, q = pos - 22 * r;
        int ay = h0 - 3 + r, ax = w0 - 3 + q;
        float v = 0.f;
        if (ay >= 0 && ay < HIMG && ax >= 0 && ax < WIMG)
            v = flow_map[(((size_t)n * HIMG + ay) * WIMG + ax) * 2 + c];
        inld[i] = v;
    }
    __syncthreads();

    // ---- conv1: 20x20 t1 positions (origin h0-2, w0-2), 16 ch, relu ----
    for (int q = tid; q < 400; q += 256) {
        int qy = q / 20, qx = q - 20 * qy;
        int ay = h0 - 2 + qy, ax = w0 - 2 + qx;
        float s[16];
        if (ay >= 0 && ay < HIMG && ax >= 0 && ax < WIMG) {
            #pragma unroll 4
            for (int co = 0; co < 16; ++co) {
                float acc = b1ld[co];
                #pragma unroll
                for (int ky = 0; ky < 3; ++ky)
                    #pragma unroll
                    for (int kx = 0; kx < 3; ++kx) {
                        const float* ip = &inld[((qy + ky) * 22 + (qx + kx)) * 2];
                        const float* wp = &w1ld[(co * 2) * 9 + ky * 3 + kx];
                        acc = fmaf(wp[0], ip[0], acc);
                        acc = fmaf(wp[9], ip[1], acc);
                    }
                s[co] = fmaxf(acc, 0.f);
            }
        } else {
            #pragma unroll
            for (int co = 0; co < 16; ++co) s[co] = 0.f;
        }
        float4* dst = (float4*)&t1ld[q * 16];       // 64B aligned
        dst[0] = make_float4(s[0],  s[1],  s[2],  s[3]);
        dst[1] = make_float4(s[4],  s[5],  s[6],  s[7]);
        dst[2] = make_float4(s[8],  s[9],  s[10], s[11]);
        dst[3] = make_float4(s[12], s[13], s[14], s[15]);
    }
    __syncthreads();

    // ---- conv2 via V_WMMA_F32_16X16X4_F32 ----
    // GEMM: t2[m=32][n=324] = W2[32][144] x im2col(t1)[144][324]
    // N tiled by 16 (21 tiles, last padded); M tiled by 16 (2 tiles);
    // K=144 in 36 unrolled steps of 4. A/B element layout: k = kbase + j + 2*(lane>=16).
    const int lane = tid & 31;
    const int wave = tid >> 5;
    const int hi   = lane >> 4;
    const int nl   = lane & 15;
    const int abase = nl * 2 + hi * 64;             // A operand per-lane base (floats)
    for (int nt = wave; nt < 21; nt += 8) {          // wave-uniform -> EXEC all-1
        int p  = nt * 16 + nl;
        int pc = p < 324 ? p : 0;
        int py = pc / 18, px = pc - 18 * py;
        int pbase = (py * 20 + px) * 16;             // t1 tile base for this pos
        v8f acc0 = {};
        v8f acc1 = {};
        #pragma unroll
        for (int ks = 0; ks < 144; ks += 4) {
            // B: two consecutive floats (k0, k0+1 share tap, cin even -> cin+1)
            int bo = hi ? boff_of(ks + 2) : boff_of(ks);   // two immediates + cndmask
            v2f B  = *(const v2f*)&t1ld[pbase + bo];
            // A: repacked pairs, one b64 per M-tile
            const float* ap = &w2ld[(ks >> 1) * 64 + abase];
            v2f A0 = *(const v2f*)ap;
            v2f A1 = *(const v2f*)(ap + 32);
            acc0 = __builtin_amdgcn_wmma_f32_16x16x4_f32(
                false, A0, false, B, (short)0, acc0, false, false);
            acc1 = __builtin_amdgcn_wmma_f32_16x16x4_f32(
                false, A1, false, B, (short)0, acc1, false, false);
        }
        if (p < 324) {
            int ay = h0 - 1 + py, ax = w0 - 1 + px;
            bool inimg = (ay >= 0 && ay < HIMG && ax >= 0 && ax < WIMG);
            float t0[8], t1v[8];
            #pragma unroll
            for (int i = 0; i < 8; ++i) {
                int ch = i + 8 * hi;                 // D layout: M = vgpr + 8*hi
                t0[i]  = inimg ? fmaxf(acc0[i] + b2ld[ch], 0.f)      : 0.f;
                t1v[i] = inimg ? fmaxf(acc1[i] + b2ld[16 + ch], 0.f) : 0.f;
            }
            float* d0 = &t2ld[p * 32 + 8 * hi];      // 16B aligned
            ((float4*)d0)[0] = make_float4(t0[0], t0[1], t0[2], t0[3]);
            ((float4*)d0)[1] = make_float4(t0[4], t0[5], t0[6], t0[7]);
            float* d1 = d0 + 16;
            ((float4*)d1)[0] = make_float4(t1v[0], t1v[1], t1v[2], t1v[3]);
            ((float4*)d1)[1] = make_float4(t1v[4], t1v[5], t1v[6], t1v[7]);
        }
    }
    __syncthreads();

    // ---- conv3 + smooth_l1 terms ----
    int ty = tid >> 4, tx = tid & 15;
    int ay = h0 + ty, ax = w0 + tx;
    float lsum = 0.f;
    if (ay < HIMG && ax < WIMG) {
        float s0 = b3ld[0], s1 = b3ld[1];
        #pragma unroll
        for (int dy = 0; dy < 3; ++dy)
            #pragma unroll
            for (int dx = 0; dx < 3; ++dx) {
                const float4* tp = (const float4*)&t2ld[((ty + dy) * 18 + (tx + dx)) * 32];
                int tap = dy * 3 + dx;
                #pragma unroll
                for (int v = 0; v < 8; ++v) {
                    float4 t = tp[v];
                    int ci = v * 4;
                    s0 = fmaf(w3ld[(ci + 0) * 9 + tap], t.x, s0);
                    s0 = fmaf(w3ld[(ci + 1) * 9 + tap], t.y, s0);
                    s0 = fmaf(w3ld[(ci + 2) * 9 + tap], t.z, s0);
                    s0 = fmaf(w3ld[(ci + 3) * 9 + tap], t.w, s0);
                    s1 = fmaf(w3ld[288 + (ci + 0) * 9 + tap], t.x, s1);
                    s1 = fmaf(w3ld[288 + (ci + 1) * 9 + tap], t.y, s1);
                    s1 = fmaf(w3ld[288 + (ci + 2) * 9 + tap], t.z, s1);
                    s1 = fmaf(w3ld[288 + (ci + 3) * 9 + tap], t.w, s1);
                }
            }
        size_t oidx = (((size_t)n * HIMG + ay) * WIMG + ax) * 2;
        flow_out[oidx]     = s0;
        flow_out[oidx + 1] = s1;
        float dgx = (float)ax - flow_gt[(((size_t)n * 2 + 0) * HIMG + ay) * WIMG + ax];
        float dgy = (float)ay - flow_gt[(((size_t)n * 2 + 1) * HIMG + ay) * WIMG + ax];
        float gtx = dgx * (2.0f / WIMG) - 1.0f;
        float gty = dgy * (2.0f / HIMG) - 1.0f;
        float d0 = s0 - gtx, d1 = s1 - gty;
        float a0 = fabsf(d0), a1 = fabsf(d1);
        lsum = (a0 < 1.f ? 0.5f * d0 * d0 : a0 - 0.5f)
             + (a1 < 1.f ? 0.5f * d1 * d1 : a1 - 0.5f);
    }
    red[tid] = lsum;
    __syncthreads();
    for (int s = 128; s > 0; s >>= 1) {
        if (tid < s) red[tid] += red[tid + s];
        __syncthreads();
    }
    if (tid == 0) atomicAdd(loss_acc, red[0]);
}

// ---------------------------------------------------------------------------
// grid_sample_nearest(latest, flow) * reserved_mask  (bandwidth kernel)
// ---------------------------------------------------------------------------
__global__ __launch_bounds__(256) void gather_kernel(
    const float* __restrict__ feats,      // (48,64,100,252); latest = k==0
    const float* __restrict__ flow,       // (16,100,252,2)
    const float* __restrict__ mask,       // (16,64,100,252)
    float* __restrict__ out, int npix)
{
    int idx = blockIdx.x * 256 + threadIdx.x;
    if (idx >= npix) return;
    int w = idx % WIMG;
    int t = idx / WIMG;
    int h = t % HIMG;
    int n = t / HIMG;
    float fx = flow[(size_t)idx * 2];
    float fy = flow[(size_t)idx * 2 + 1];
    float ix = (fx + 1.f) * (0.5f * WIMG) - 0.5f;
    float iy = (fy + 1.f) * (0.5f * HIMG) - 0.5f;
    float ixr = rintf(ix);                // round-half-even == jnp.round
    float iyr = rintf(iy);
    float vmul = (ixr >= 0.f && ixr <= (float)(WIMG - 1) &&
                  iyr >= 0.f && iyr <= (float)(HIMG - 1)) ? 1.f : 0.f;
    int ixc = (int)fminf(fmaxf(ixr, 0.f), (float)(WIMG - 1));
    int iyc = (int)fminf(fmaxf(iyr, 0.f), (float)(HIMG - 1));
    const size_t plane = (size_t)HIMG * WIMG;
    const float* src = feats + (size_t)(n * 3) * CFEAT * plane + (size_t)iyc * WIMG + ixc;
    const float* msk = mask  + (size_t)n * CFEAT * plane + (size_t)h * WIMG + w;
    float*       dst = out   + (size_t)n * CFEAT * plane + (size_t)h * WIMG + w;
    #pragma unroll 4
    for (int c = 0; c < CFEAT; ++c)
        dst[c * plane] = src[c * plane] * msk[c * plane] * vmul;
}

__global__ void finalize_kernel(const float* __restrict__ loss_acc,
                                float* __restrict__ out_loss)
{
    *out_loss = *loss_acc * (1.0f / (float)(NIMG * HIMG * WIMG * 2));
}

// ---------------------------------------------------------------------------
extern "C" void kernel_launch(void* const* d_in, const int* in_sizes, int n_in,
                              void* d_out, int out_size, void* d_ws, size_t ws_size,
                              hipStream_t stream) {
    const float* feats    = (const float*)d_in[0];
    // d_in[1] pairwise_t_matrix, d_in[2] record_len: only shapes used by ref
    const float* flow_map = (const float*)d_in[3];
    const float* mask     = (const float*)d_in[4];
    const float* flow_gt  = (const float*)d_in[5];
    const float* w1 = (const float*)d_in[6];
    const float* b1 = (const float*)d_in[7];
    const float* w2 = (const float*)d_in[8];
    const float* b2 = (const float*)d_in[9];
    const float* w3 = (const float*)d_in[10];
    const float* b3 = (const float*)d_in[11];

    float* out      = (float*)d_out;
    float* flow_buf = (float*)d_ws;                              // 16*100*252*2 floats
    float* loss_acc = flow_buf + (size_t)NIMG * HIMG * WIMG * 2; // 1 float

    zero_loss_kernel<<<1, 1, 0, stream>>>(loss_acc);

    dim3 grid((WIMG + TILE - 1) / TILE,   // 16
              (HIMG + TILE - 1) / TILE,   // 7
              NIMG);                      // 16
    size_t shbytes = (4608 + 6400 + 10752 + 968 + 288 + 16 + 32 + 576 + 2 + 256) * 4; // ~94 KB
    flow_conv_kernel<<<grid, 256, shbytes, stream>>>(
        flow_map, flow_gt, w1, b1, w2, b2, w3, b3, flow_buf, loss_acc);

    int npix = NIMG * HIMG * WIMG;
    gather_kernel<<<(npix + 255) / 256, 256, 0, stream>>>(feats, flow_buf, mask, out, npix);

    finalize_kernel<<<1, 1, 0, stream>>>(loss_acc, out + (size_t)npix * CFEAT);
}